// GraphNet_19619410608795
// MI455X (gfx1250) — compile-verified
//
#include <hip/hip_runtime.h>
#include <float.h>
#include <math.h>

typedef __attribute__((ext_vector_type(2))) float v2f;
typedef __attribute__((ext_vector_type(8))) float v8f;

#define NEDGE   200000
#define NGRAPH  50
#define NPAT    10
#define FEATD   1036
#define NHID    128
#define KC      64        // K-chunk staged in LDS
#define AST     66        // LDS row stride (bank-conflict padding, even for b64 reads)

// ------------------------------ utility kernels ------------------------------

__global__ void k_fill_f32(float* __restrict__ p, float v, int n) {
  int i = blockIdx.x * blockDim.x + threadIdx.x;
  if (i < n) p[i] = v;
}

__global__ void k_fill_i32(int* __restrict__ p, int v, int n) {
  int i = blockIdx.x * blockDim.x + threadIdx.x;
  if (i < n) p[i] = v;
}

__global__ void k_edge_init(const int* __restrict__ ei, int* __restrict__ esrc,
                            int* __restrict__ edst, float* __restrict__ em) {
  int e = blockIdx.x * blockDim.x + threadIdx.x;
  if (e < NEDGE) { esrc[e] = ei[e]; edst[e] = ei[NEDGE + e]; em[e] = 1.0f; }
}

// max of first 12 feature columns over all rows (values in [0,1): int-bit
// atomicMax on positive float bits preserves float ordering)
__global__ void k_colmax(const float* __restrict__ x, float* __restrict__ colmax, int rows) {
  __shared__ float red[256];
  float loc[12];
  #pragma unroll
  for (int c = 0; c < 12; ++c) loc[c] = 0.0f;
  for (int r = blockIdx.x * blockDim.x + threadIdx.x; r < rows; r += gridDim.x * blockDim.x) {
    const float* row = x + (size_t)r * FEATD;
    #pragma unroll
    for (int c = 0; c < 12; ++c) loc[c] = fmaxf(loc[c], row[c]);
  }
  for (int c = 0; c < 12; ++c) {
    red[threadIdx.x] = loc[c];
    __syncthreads();
    for (int o = 128; o > 0; o >>= 1) {
      if (threadIdx.x < o) red[threadIdx.x] = fmaxf(red[threadIdx.x], red[threadIdx.x + o]);
      __syncthreads();
    }
    if (threadIdx.x == 0) atomicMax((int*)&colmax[c], __float_as_int(red[0]));
    __syncthreads();
  }
}

__global__ void k_build_scale(const float* __restrict__ colmax, float* __restrict__ scale) {
  int i = blockIdx.x * blockDim.x + threadIdx.x;
  if (i < FEATD) scale[i] = (i < 12) ? 1.0f / colmax[i] : 1.0f;
}

// ------------------------------ fused dual WMMA fp32 GEMM ------------------------------
// CL = (A .* colscale) @ WL ; CR = (A .* colscale) @ WR  in one pass over A.
// Block: 256 threads = 8 waves. Block owns 16 rows x all 128 cols; wave w owns
// column tile [16w,16w+16) for BOTH weight matrices. A chunk (16 x kc, kc<=64)
// is cooperatively staged in LDS (scale folded in), then each wave runs kc/4
// V_WMMA_F32_16X16X4_F32 steps per matrix. A is read from HBM exactly once.
// A fragment (16x4 f32): lanes 0-15 -> K=k0,k0+1 ; lanes 16-31 -> K=k0+2,k0+3.
// C fragment: VGPR v -> row tm + v + 8*(lane>=16), col tn + (lane&15).
template <bool SC>
__global__ void k_gemm_dual_wmma(const float* __restrict__ A, int lda, int K,
                                 const float* __restrict__ WL,
                                 const float* __restrict__ WR,
                                 const float* __restrict__ scale,
                                 float* __restrict__ CL, float* __restrict__ CR) {
  __shared__ float As[16 * AST];
  const int t    = threadIdx.x;          // 0..255
  const int wave = t >> 5;               // 0..7 -> n-tile
  const int lane = t & 31;
  const int half = lane >> 4;
  const int l    = lane & 15;
  const int tm   = blockIdx.x << 4;
  const int tn   = wave << 4;
  const int sr   = t >> 4;               // staging row 0..15
  const int sc0  = t & 15;               // staging col start
  const float* arow = A + (size_t)(tm + sr) * lda;
  v8f accL = {}, accR = {};
  for (int k0 = 0; k0 < K; k0 += KC) {
    const int kc = min(KC, K - k0);
    __syncthreads();
    for (int c = sc0; c < kc; c += 16) {
      float v = arow[k0 + c];
      if (SC) v *= scale[k0 + c];
      As[sr * AST + c] = v;
    }
    __syncthreads();
    for (int ks = 0; ks < kc; ks += 4) {
      const int kk = ks + 2 * half;
      v2f av;
      av.x = As[l * AST + kk];
      av.y = As[l * AST + kk + 1];
      const float* wl0 = WL + (size_t)(k0 + kk) * NHID + tn + l;
      const float* wr0 = WR + (size_t)(k0 + kk) * NHID + tn + l;
      v2f bL; bL.x = wl0[0]; bL.y = wl0[NHID];
      v2f bR; bR.x = wr0[0]; bR.y = wr0[NHID];
      accL = __builtin_amdgcn_wmma_f32_16x16x4_f32(false, av, false, bL,
                                                   (short)0, accL, false, false);
      accR = __builtin_amdgcn_wmma_f32_16x16x4_f32(false, av, false, bR,
                                                   (short)0, accR, false, false);
    }
  }
  const int row = tm + half * 8;
  const int col = tn + l;
  #pragma unroll
  for (int v = 0; v < 8; ++v) {
    CL[(size_t)(row + v) * NHID + col] = accL[v];
    CR[(size_t)(row + v) * NHID + col] = accR[v];
  }
}

// ------------------------------ graph plumbing ------------------------------

// agg[dst] += yl[src] * m   (128-dim, 4 floats per thread)
__global__ void k_scatter_agg(const float* __restrict__ yl, const int* __restrict__ esrc,
                              const int* __restrict__ edst, const float* __restrict__ em,
                              float* __restrict__ agg) {
  int t = blockIdx.x * blockDim.x + threadIdx.x;
  if (t >= NEDGE * 32) return;
  int e = t >> 5, q = (t & 31) << 2;
  float m = em[e];
  if (m == 0.0f) return;
  const float* ys = yl + (size_t)esrc[e] * NHID + q;
  float* ad = agg + (size_t)edst[e] * NHID + q;
  #pragma unroll
  for (int j = 0; j < 4; ++j) atomicAdd(ad + j, ys[j] * m);
}

__global__ void k_scatter_cnt(const int* __restrict__ edst, const float* __restrict__ em,
                              float* __restrict__ cnt) {
  int e = blockIdx.x * blockDim.x + threadIdx.x;
  if (e < NEDGE) { float m = em[e]; if (m != 0.0f) atomicAdd(&cnt[edst[e]], m); }
}

// xs = relu(agg / max(cnt,1) + yr + bl)
__global__ void k_sage_finish(const float* __restrict__ agg, const float* __restrict__ cnt,
                              const float* __restrict__ yr, const float* __restrict__ bl,
                              float* __restrict__ xs, int M) {
  int t = blockIdx.x * blockDim.x + threadIdx.x;
  if (t >= M * NHID) return;
  int i = t >> 7, f = t & 127;
  float v = agg[t] / fmaxf(cnt[i], 1.0f) + yr[t] + bl[f];
  xs[t] = v > 0.0f ? v : 0.0f;
}

// per-node dot products with wrel / wroot (one wave per node)
__global__ void k_score_dots(const float* __restrict__ x, const float* __restrict__ wrel,
                             const float* __restrict__ wroot, float* __restrict__ srel,
                             float* __restrict__ sroot, int M) {
  int gt = blockIdx.x * blockDim.x + threadIdx.x;
  int wid = gt >> 5, lane = gt & 31;
  if (wid >= M) return;
  const float* row = x + (size_t)wid * NHID;
  float a = 0.f, b = 0.f;
  for (int j = lane; j < NHID; j += 32) { float v = row[j]; a += v * wrel[j]; b += v * wroot[j]; }
  for (int o = 16; o > 0; o >>= 1) { a += __shfl_down(a, o, 32); b += __shfl_down(b, o, 32); }
  if (lane == 0) { srel[wid] = a; sroot[wid] = b; }
}

__global__ void k_score_scatter(const float* __restrict__ srel, const int* __restrict__ esrc,
                                const int* __restrict__ edst, const float* __restrict__ em,
                                float* __restrict__ sagg) {
  int e = blockIdx.x * blockDim.x + threadIdx.x;
  if (e < NEDGE) { float m = em[e]; if (m != 0.0f) atomicAdd(&sagg[edst[e]], srel[esrc[e]] * m); }
}

__global__ void k_score_finish(float* __restrict__ sagg, const float* __restrict__ sroot,
                               const float* __restrict__ brel, int M) {
  int i = blockIdx.x * blockDim.x + threadIdx.x;
  if (i < M) sagg[i] += sroot[i] + brel[0];
}

// per-graph top-k by iterative argmax in LDS; writes perm (old global ids),
// sel (scores), mapping (old id -> new id, -1 if dropped)
__global__ void k_topk(const float* __restrict__ score, int n, int k,
                       int* __restrict__ perm, float* __restrict__ sel,
                       int* __restrict__ mapping) {
  int g = blockIdx.x, t = threadIdx.x;
  __shared__ float s[1024];
  __shared__ float rv[256];
  __shared__ int   ri[256];
  for (int i = t; i < n; i += 256) s[i] = score[g * n + i];
  __syncthreads();
  for (int it = 0; it < k; ++it) {
    float bv = -FLT_MAX; int bi = -1;
    for (int i = t; i < n; i += 256) if (s[i] > bv) { bv = s[i]; bi = i; }
    rv[t] = bv; ri[t] = bi;
    __syncthreads();
    for (int o = 128; o > 0; o >>= 1) {
      if (t < o && rv[t + o] > rv[t]) { rv[t] = rv[t + o]; ri[t] = ri[t + o]; }
      __syncthreads();
    }
    if (t == 0) {
      int idx = ri[0];
      sel[g * k + it]  = rv[0];
      perm[g * k + it] = g * n + idx;
      mapping[g * n + idx] = g * k + it;
      s[idx] = -FLT_MAX;
    }
    __syncthreads();
  }
}

// xnext[j] = xs[perm[j]] * tanh(sel[j])
__global__ void k_gather_gate(const float* __restrict__ xs, const int* __restrict__ perm,
                              const float* __restrict__ sel, float* __restrict__ xnext, int Mk) {
  int t = blockIdx.x * blockDim.x + threadIdx.x;
  if (t >= Mk * NHID) return;
  int j = t >> 7, f = t & 127;
  xnext[t] = xs[(size_t)perm[j] * NHID + f] * tanhf(sel[j]);
}

__global__ void k_remap(int* __restrict__ esrc, int* __restrict__ edst,
                        float* __restrict__ em, const int* __restrict__ mapping) {
  int e = blockIdx.x * blockDim.x + threadIdx.x;
  if (e >= NEDGE) return;
  int ns = mapping[esrc[e]], nd = mapping[edst[e]];
  float m = em[e];
  em[e]   = (ns >= 0 && nd >= 0) ? m : 0.0f;
  esrc[e] = ns > 0 ? ns : 0;
  edst[e] = nd > 0 ? nd : 0;
}

// h[g] += concat(max_i xk[g,i,:], mean_i xk[g,i,:])
__global__ void k_readout(const float* __restrict__ xk, int k, float* __restrict__ h) {
  int g = blockIdx.x, f = threadIdx.x;   // 128 threads
  float mx = -FLT_MAX, sm = 0.f;
  for (int i = 0; i < k; ++i) {
    float v = xk[((size_t)(g * k + i)) * NHID + f];
    mx = fmaxf(mx, v); sm += v;
  }
  h[g * 256 + f]       += mx;
  h[g * 256 + 128 + f] += sm / (float)k;
}

// ------------------------------ MLP head ------------------------------

__global__ void k_head(const float* __restrict__ h, const int* __restrict__ gpp,
                       const float* __restrict__ w1, const float* __restrict__ b1,
                       const float* __restrict__ w2, const float* __restrict__ b2,
                       const float* __restrict__ gw, const float* __restrict__ gb,
                       const float* __restrict__ hw, const float* __restrict__ hb,
                       float* __restrict__ out) {
  __shared__ float pooled[NPAT * 256];
  __shared__ float h1[NPAT * 128];
  __shared__ float ft[NPAT * 32];
  int t = threadIdx.x;
  for (int idx = t; idx < NPAT * 256; idx += blockDim.x) {
    int p = idx >> 8, c = idx & 255;
    int start = 0;
    for (int q = 0; q < p; ++q) start += gpp[q];
    int cnt = gpp[p];
    float s = 0.f;
    for (int g = 0; g < cnt; ++g) s += h[(start + g) * 256 + c];
    pooled[idx] = s / (float)cnt;
  }
  __syncthreads();
  for (int idx = t; idx < NPAT * 128; idx += blockDim.x) {
    int p = idx >> 7, j = idx & 127;
    float s = b1[j];
    for (int c = 0; c < 256; ++c) s += pooled[p * 256 + c] * w1[c * 128 + j];
    h1[idx] = s > 0.f ? s : 0.f;
  }
  __syncthreads();
  for (int idx = t; idx < NPAT * 32; idx += blockDim.x) {
    int p = idx >> 5, j = idx & 31;
    float s = b2[j];
    for (int c = 0; c < 128; ++c) s += h1[p * 128 + c] * w2[c * 32 + j];
    float v = s > 0.f ? s : 0.f;
    ft[idx] = v;
    out[idx] = v;                          // feats -> out[0:320)
  }
  __syncthreads();
  if (t < NPAT) {
    int p = t;
    float z[3];
    for (int j = 0; j < 3; ++j) {
      float s = gb[j];
      for (int c = 0; c < 32; ++c) s += ft[p * 32 + c] * gw[c * 3 + j];
      z[j] = s;
    }
    float mz = fmaxf(z[0], fmaxf(z[1], z[2]));
    float ls = mz + logf(expf(z[0] - mz) + expf(z[1] - mz) + expf(z[2] - mz));
    for (int j = 0; j < 3; ++j) out[NPAT * 32 + p * 3 + j] = z[j] - ls;   // grade
    float hz = hb[0];
    for (int c = 0; c < 32; ++c) hz += ft[p * 32 + c] * hw[c];
    hz = 1.0f / (1.0f + expf(-hz));
    out[NPAT * 32 + NPAT * 3 + p] = hz * 6.0f - 3.0f;                     // hazard
  }
}

// ------------------------------ host ------------------------------

extern "C" void kernel_launch(void* const* d_in, const int* in_sizes, int n_in,
                              void* d_out, int out_size, void* d_ws, size_t ws_size,
                              hipStream_t stream) {
  (void)in_sizes; (void)n_in; (void)out_size; (void)ws_size;
  const float* x   = (const float*)d_in[0];
  const int*   ei  = (const int*)d_in[1];
  const int*   gpp = (const int*)d_in[2];
  const float* c_wl[3]   = {(const float*)d_in[3],  (const float*)d_in[9],  (const float*)d_in[15]};
  const float* c_bl[3]   = {(const float*)d_in[4],  (const float*)d_in[10], (const float*)d_in[16]};
  const float* c_wr[3]   = {(const float*)d_in[5],  (const float*)d_in[11], (const float*)d_in[17]};
  const float* p_wrel[3] = {(const float*)d_in[6],  (const float*)d_in[12], (const float*)d_in[18]};
  const float* p_brel[3] = {(const float*)d_in[7],  (const float*)d_in[13], (const float*)d_in[19]};
  const float* p_wroot[3]= {(const float*)d_in[8],  (const float*)d_in[14], (const float*)d_in[20]};
  const float* lin1_w = (const float*)d_in[21]; const float* lin1_b = (const float*)d_in[22];
  const float* lin2_w = (const float*)d_in[23]; const float* lin2_b = (const float*)d_in[24];
  const float* grade_w= (const float*)d_in[25]; const float* grade_b= (const float*)d_in[26];
  const float* haz_w  = (const float*)d_in[27]; const float* haz_b  = (const float*)d_in[28];
  float* out = (float*)d_out;

  // workspace carve-out
  char* wp = (char*)d_ws;
  auto carve = [&](size_t bytes) -> void* {
    void* p = (void*)wp; wp += (bytes + 255) & ~(size_t)255; return p;
  };
  const int MMAX = NGRAPH * 1000;          // 50000
  float* colmax = (float*)carve(12 * 4);
  float* scale  = (float*)carve(FEATD * 4);
  float* yl     = (float*)carve((size_t)MMAX * NHID * 4);
  float* yr     = (float*)carve((size_t)MMAX * NHID * 4);
  float* agg    = (float*)carve((size_t)MMAX * NHID * 4);
  float* xs     = (float*)carve((size_t)MMAX * NHID * 4);
  float* xpA    = (float*)carve((size_t)10000 * NHID * 4);
  float* xpB    = (float*)carve((size_t)10000 * NHID * 4);
  float* cnt    = (float*)carve((size_t)MMAX * 4);
  float* srel   = (float*)carve((size_t)MMAX * 4);
  float* sroot  = (float*)carve((size_t)MMAX * 4);
  float* sagg   = (float*)carve((size_t)MMAX * 4);
  int*   mapping= (int*)  carve((size_t)MMAX * 4);
  int*   perm   = (int*)  carve((size_t)10000 * 4);
  float* sel    = (float*)carve((size_t)10000 * 4);
  int*   esrc   = (int*)  carve((size_t)NEDGE * 4);
  int*   edst   = (int*)  carve((size_t)NEDGE * 4);
  float* em     = (float*)carve((size_t)NEDGE * 4);
  float* hacc   = (float*)carve((size_t)NGRAPH * 256 * 4);

  auto nb = [](int n, int b) { return (n + b - 1) / b; };

  // normalization scale + edge state + readout accumulator
  k_fill_f32<<<1, 32, 0, stream>>>(colmax, 0.0f, 12);
  k_colmax<<<128, 256, 0, stream>>>(x, colmax, MMAX);
  k_build_scale<<<nb(FEATD, 256), 256, 0, stream>>>(colmax, scale);
  k_edge_init<<<nb(NEDGE, 256), 256, 0, stream>>>(ei, esrc, edst, em);
  k_fill_f32<<<nb(NGRAPH * 256, 256), 256, 0, stream>>>(hacc, 0.0f, NGRAPH * 256);

  const int ns[3] = {1000, 200, 40};
  const int ks[3] = {200, 40, 8};
  const float* xin = x;
  int lda = FEATD, K = FEATD;
  float* xbuf[2] = {xpA, xpB};

  for (int l = 0; l < 3; ++l) {
    const int n = ns[l], k = ks[l];
    const int M = NGRAPH * n, Mk = NGRAPH * k;

    // SAGE: yl = xin@wl (scatter-mean applied after), yr = xin@wr, fused single
    // pass over xin with LDS-staged A tiles.
    if (l == 0) {
      k_gemm_dual_wmma<true><<<M / 16, 256, 0, stream>>>(xin, lda, K, c_wl[l], c_wr[l],
                                                         scale, yl, yr);
    } else {
      k_gemm_dual_wmma<false><<<M / 16, 256, 0, stream>>>(xin, lda, K, c_wl[l], c_wr[l],
                                                          nullptr, yl, yr);
    }
    k_fill_f32<<<nb(M * NHID, 256), 256, 0, stream>>>(agg, 0.0f, M * NHID);
    k_fill_f32<<<nb(M, 256), 256, 0, stream>>>(cnt, 0.0f, M);
    k_scatter_agg<<<nb(NEDGE * 32, 256), 256, 0, stream>>>(yl, esrc, edst, em, agg);
    k_scatter_cnt<<<nb(NEDGE, 256), 256, 0, stream>>>(edst, em, cnt);
    k_sage_finish<<<nb(M * NHID, 256), 256, 0, stream>>>(agg, cnt, yr, c_bl[l], xs, M);

    // SAGPool
    k_score_dots<<<nb(M * 32, 256), 256, 0, stream>>>(xs, p_wrel[l], p_wroot[l], srel, sroot, M);
    k_fill_f32<<<nb(M, 256), 256, 0, stream>>>(sagg, 0.0f, M);
    k_score_scatter<<<nb(NEDGE, 256), 256, 0, stream>>>(srel, esrc, edst, em, sagg);
    k_score_finish<<<nb(M, 256), 256, 0, stream>>>(sagg, sroot, p_brel[l], M);
    k_fill_i32<<<nb(M, 256), 256, 0, stream>>>(mapping, -1, M);
    k_topk<<<NGRAPH, 256, 0, stream>>>(sagg, n, k, perm, sel, mapping);

    float* xnext = xbuf[l & 1];
    k_gather_gate<<<nb(Mk * NHID, 256), 256, 0, stream>>>(xs, perm, sel, xnext, Mk);
    k_remap<<<nb(NEDGE, 256), 256, 0, stream>>>(esrc, edst, em, mapping);
    k_readout<<<NGRAPH, 128, 0, stream>>>(xnext, k, hacc);

    xin = xnext; lda = NHID; K = NHID;
  }

  k_head<<<1, 256, 0, stream>>>(hacc, gpp, lin1_w, lin1_b, lin2_w, lin2_b,
                                grade_w, grade_b, haz_w, haz_b, out);
}